// ImprovedChamferDistanceLoss_75299366633597
// MI455X (gfx1250) — compile-verified
//
#include <hip/hip_runtime.h>
#include <math.h>

// Chamfer distance, B=8 T=8 N=2048 C=3.
// cross[n,m] = p[n]·q[m] via V_WMMA_F32_16X16X4_F32 (K=4, 4th coord zero-padded).
// Identity used:  max(0, min_m(p2 + q2_m - 2c_m)) = max(0, p2 + min_m(q2_m - 2c_m))
// so the inner loop only tracks min_m(q2_m - 2c_m); p2 and the clamp are applied once.

typedef float v2f __attribute__((ext_vector_type(2)));
typedef float v8f __attribute__((ext_vector_type(8)));

#define CH_EPS 1e-8f

__global__ void chamfer_init_out(float* out) {
    if (threadIdx.x == 0 && blockIdx.x == 0) out[0] = 0.0f;
}

// One wave handles TWO 16-row P tiles (dir, bt, rowPair) and sweeps all N Q
// points in 16-wide column tiles; each B tile load feeds two WMMAs.
__global__ __launch_bounds__(256)
void ImprovedChamferDistanceLoss_75299366633597_kernel(
    const float* __restrict__ pred,
    const float* __restrict__ tgt,
    float* __restrict__ out,
    int BT, int N)
{
    const int lane        = threadIdx.x & 31;
    const int waveInBlock = threadIdx.x >> 5;
    const int wid         = blockIdx.x * (blockDim.x >> 5) + waveInBlock;

    const int pairsPerBT  = N >> 5;              // 64 (two 16-row tiles each)
    const int wavesPerDir = BT * pairsPerBT;     // 4096

    const int dir = (wid >= wavesPerDir) ? 1 : 0;
    int rem       = wid - dir * wavesPerDir;
    const int bt      = rem / pairsPerBT;
    const int rowPair = rem - bt * pairsPerBT;

    const float* __restrict__ P = dir ? tgt : pred;
    const float* __restrict__ Q = dir ? pred : tgt;
    const float* Pbase = P + (size_t)bt * N * 3;
    const float* Qbase = Q + (size_t)bt * N * 3;

    const int hi  = lane >> 4;     // 0: K=0,1 half   1: K=2,3 half
    const int l15 = lane & 15;

    // ---- Two A tiles (16 rows x K4, K=3 padded). ISA layout:
    // lanes 0-15: VGPR0=K0, VGPR1=K1 ; lanes 16-31: VGPR0=K2, VGPR1=K3(=0)
    const float* prow0 = Pbase + (size_t)(rowPair * 32 + l15) * 3;
    const float* prow1 = prow0 + 48;   // next 16 points
    const float p0x = prow0[0], p0y = prow0[1], p0z = prow0[2];
    const float p1x = prow1[0], p1y = prow1[1], p1z = prow1[2];
    v2f a0, a1;
    a0.x = hi ? p0z : p0x;   a0.y = hi ? 0.0f : p0y;
    a1.x = hi ? p1z : p1x;   a1.y = hi ? 0.0f : p1y;

    const float p2_own0 = p0x*p0x + p0y*p0y + p0z*p0z;
    const float p2_own1 = p1x*p1x + p1y*p1y + p1z*p1z;

    // Running min_m(q2_m - 2*cross) per owned row; p2 added after the loop.
    float rm0[8], rm1[8];
#pragma unroll
    for (int v = 0; v < 8; ++v) { rm0[v] = 3.0e38f; rm1[v] = 3.0e38f; }

    const int tilesPerBT = N >> 4;   // 128 column tiles
    for (int j = 0; j < tilesPerBT; ++j) {
        // ---- B tile: K4 x 16 cols, lane = column. Same half-split as A.
        const float* qrow = Qbase + (size_t)(j * 16 + l15) * 3;
        const float qx = qrow[0], qy = qrow[1], qz = qrow[2];
        const float q2 = qx*qx + qy*qy + qz*qz;
        v2f b;
        b.x = hi ? qz   : qx;
        b.y = hi ? 0.0f : qy;

        v8f c0 = {}, c1 = {};
        c0 = __builtin_amdgcn_wmma_f32_16x16x4_f32(
                 false, a0, false, b, (short)0, c0, false, false);
        c1 = __builtin_amdgcn_wmma_f32_16x16x4_f32(
                 false, a1, false, b, (short)0, c1, false, false);

        // C/D layout: VGPR v -> row (v + 8*hi), lane -> col l15
#pragma unroll
        for (int v = 0; v < 8; ++v) {
            rm0[v] = fminf(rm0[v], fmaf(-2.0f, c0[v], q2));
            rm1[v] = fminf(rm1[v], fmaf(-2.0f, c1[v], q2));
        }
    }

    // Min over the 16 columns held by each half-wave (xor stays within half)
#pragma unroll
    for (int m = 1; m <= 8; m <<= 1) {
#pragma unroll
        for (int v = 0; v < 8; ++v) {
            rm0[v] = fminf(rm0[v], __shfl_xor(rm0[v], m, 32));
            rm1[v] = fminf(rm1[v], __shfl_xor(rm1[v], m, 32));
        }
    }

    // Each lane needs p2 of rows hi*8 + v for both tiles
    float s = 0.0f;
#pragma unroll
    for (int v = 0; v < 8; ++v) {
        const float p2v0 = __shfl(p2_own0, hi * 8 + v, 32);
        const float p2v1 = __shfl(p2_own1, hi * 8 + v, 32);
        const float d0 = fmaxf(rm0[v] + p2v0, 0.0f);
        const float d1 = fmaxf(rm1[v] + p2v1, 0.0f);
        s += sqrtf(d0 + CH_EPS) + sqrtf(d1 + CH_EPS);
    }
    s += __shfl_xor(s, 16, 32);       // combine halves: all 32 rows of the pair

    if (lane == 0) {
        atomicAdd(out, s / (float)(BT * N));   // mean contribution, both dirs
    }
}

extern "C" void kernel_launch(void* const* d_in, const int* in_sizes, int n_in,
                              void* d_out, int out_size, void* d_ws, size_t ws_size,
                              hipStream_t stream) {
    const float* pred = (const float*)d_in[0];
    const float* tgt  = (const float*)d_in[1];
    float* out = (float*)d_out;

    // B=8, T=8, N=2048, C=3  (in_sizes[0] == 8*8*2048*3 == 393216)
    const int BT = 64;
    const int N  = 2048;

    chamfer_init_out<<<1, 32, 0, stream>>>(out);

    const int totalWaves = 2 * BT * (N / 32);   // 8192 (each wave: 2 row tiles)
    const int wavesPerBlock = 8;                // 256 threads
    const int blocks = totalWaves / wavesPerBlock;
    ImprovedChamferDistanceLoss_75299366633597_kernel<<<blocks, 256, 0, stream>>>(
        pred, tgt, out, BT, N);
}